// VGAEModel_45286135169739
// MI455X (gfx1250) — compile-verified
//
#include <hip/hip_runtime.h>

typedef float v2f __attribute__((ext_vector_type(2)));
typedef float v8f __attribute__((ext_vector_type(8)));

#define NROWS 100000
#define MTILES 6250   // 100000 / 16, exact

// ---------------------------------------------------------------------------
// Dense GEMM  C[M, NT*16] = A[M, K] @ W[K, NT*16]  using V_WMMA_F32_16X16X4_F32.
// One wave per 16-row M-tile; iterates K in steps of 4, accumulating NT 16x16
// f32 tiles in registers (A fragment reused across the whole N strip).
// ---------------------------------------------------------------------------
template <int K, int NT>
__global__ void __launch_bounds__(128)
gemm_wmma_kernel(const float* __restrict__ A, const float* __restrict__ W,
                 float* __restrict__ C) {
    constexpr int NOUT = NT * 16;
    const int wave = threadIdx.x >> 5;
    const int lane = threadIdx.x & 31;
    const int mtile = blockIdx.x * 4 + wave;
    if (mtile >= MTILES) return;          // wave-uniform: EXEC stays all-1s

    const int l  = lane & 15;
    const int hi = lane >> 4;

    const float* Arow = A + (size_t)(mtile * 16 + l) * K + 2 * hi;

    const v8f vzero = {};
    v8f acc[NT];
#pragma unroll
    for (int nt = 0; nt < NT; ++nt) acc[nt] = vzero;

    for (int k0 = 0; k0 < K; k0 += 4) {
        // A fragment: lane l rows, K = k0 + 2*hi + {0,1}
        v2f a = *(const v2f*)(Arow + k0);
        const float* Wp = W + (size_t)(k0 + 2 * hi) * NOUT + l;
#pragma unroll
        for (int nt = 0; nt < NT; ++nt) {
            v2f b;
            b.x = Wp[nt * 16];            // W[k0+2*hi    ][n0+l]
            b.y = Wp[nt * 16 + NOUT];     // W[k0+2*hi + 1][n0+l]
            acc[nt] = __builtin_amdgcn_wmma_f32_16x16x4_f32(
                false, a, false, b, (short)0, acc[nt], false, false);
        }
    }

    // C/D layout: VGPR j -> row (j + 8*hi), lane l -> column n0+l
    float* Crow = C + (size_t)(mtile * 16 + 8 * hi) * NOUT + l;
#pragma unroll
    for (int nt = 0; nt < NT; ++nt) {
#pragma unroll
        for (int j = 0; j < 8; ++j) {
            Crow[(size_t)j * NOUT + nt * 16] = acc[nt][j];
        }
    }
}

// ---------------------------------------------------------------------------
// SpMM: out[row[e], :] += w[e] * support[col[e], :]
// One thread per (edge, 4-feature chunk); float4 gather + 4 HW f32 atomics.
// support fits in the 192MB L2, so gathers and atomics stay on-chip.
// ---------------------------------------------------------------------------
template <int F>
__global__ void __launch_bounds__(256)
spmm_kernel(const int* __restrict__ rows, const int* __restrict__ cols,
            const float* __restrict__ w, const float* __restrict__ support,
            float* __restrict__ out, int E) {
    constexpr int F4  = F / 4;        // chunks per edge
    constexpr int EPB = 256 / F4;     // edges per block
    const int e = blockIdx.x * EPB + (int)(threadIdx.x / F4);
    if (e >= E) return;
    const int c = ((int)threadIdx.x % F4) * 4;

    const int   r   = rows[e];
    const int   col = cols[e];
    const float ww  = w[e];

    const float4 v = *(const float4*)(support + (size_t)col * F + c);
    float* o = out + (size_t)r * F + c;
    unsafeAtomicAdd(o + 0, v.x * ww);
    unsafeAtomicAdd(o + 1, v.y * ww);
    unsafeAtomicAdd(o + 2, v.z * ww);
    unsafeAtomicAdd(o + 3, v.w * ww);
}

// ---------------------------------------------------------------------------
// In-place ReLU, float4 vectorized.
// ---------------------------------------------------------------------------
__global__ void __launch_bounds__(256)
relu_kernel(float* __restrict__ x, long long n4) {
    long long i = (long long)blockIdx.x * blockDim.x + threadIdx.x;
    if (i >= n4) return;
    float4* p = (float4*)x + i;
    float4 v = *p;
    v.x = fmaxf(v.x, 0.0f);
    v.y = fmaxf(v.y, 0.0f);
    v.z = fmaxf(v.z, 0.0f);
    v.w = fmaxf(v.w, 0.0f);
    *p = v;
}

extern "C" void kernel_launch(void* const* d_in, const int* in_sizes, int n_in,
                              void* d_out, int out_size, void* d_ws, size_t ws_size,
                              hipStream_t stream) {
    (void)n_in; (void)out_size; (void)ws_size;

    const float* x    = (const float*)d_in[0];
    const int*   erow = (const int*)d_in[1];
    const int*   ecol = (const int*)d_in[2];
    const float* ew   = (const float*)d_in[3];
    const float* W1   = (const float*)d_in[4];
    const float* W2   = (const float*)d_in[5];
    const float* W3   = (const float*)d_in[6];
    const int E = in_sizes[1];

    float* out    = (float*)d_out;
    float* z      = out;                                  // [N,128]
    float* mu     = out + (size_t)NROWS * 128;            // [N,128]
    float* logvar = out + (size_t)NROWS * 256;            // [N,128]

    float* bufA = (float*)d_ws;                           // support buf [N,256]
    float* bufB = bufA + (size_t)NROWS * 256;             // h1 buf      [N,256]

    // --- Layer 1 ---------------------------------------------------------
    hipMemsetAsync(bufB, 0, (size_t)NROWS * 256 * sizeof(float), stream);

    // support1 = x @ W1   (K=512, NOUT=256)
    gemm_wmma_kernel<512, 16><<<(MTILES + 3) / 4, 128, 0, stream>>>(x, W1, bufA);

    // h1_pre = spmm(support1)   (F=256 -> 4 edges/block)
    spmm_kernel<256><<<(E + 3) / 4, 256, 0, stream>>>(erow, ecol, ew, bufA, bufB, E);

    // h1 = relu(h1_pre)
    {
        long long n4 = (long long)NROWS * 256 / 4;
        relu_kernel<<<(unsigned)((n4 + 255) / 256), 256, 0, stream>>>(bufB, n4);
    }

    // --- Layer 2 / 3 -----------------------------------------------------
    // support2 = h1 @ W2 ; support3 = h1 @ W3   (K=256, NOUT=128), reuse bufA
    gemm_wmma_kernel<256, 8><<<(MTILES + 3) / 4, 128, 0, stream>>>(bufB, W2, bufA);
    gemm_wmma_kernel<256, 8><<<(MTILES + 3) / 4, 128, 0, stream>>>(
        bufB, W3, bufA + (size_t)NROWS * 128);

    // zero mu + logvar (contiguous) before atomic accumulation
    hipMemsetAsync(mu, 0, (size_t)NROWS * 256 * sizeof(float), stream);

    // mu = spmm(support2) ; logvar = spmm(support3)   (F=128 -> 8 edges/block)
    spmm_kernel<128><<<(E + 7) / 8, 256, 0, stream>>>(erow, ecol, ew, bufA, mu, E);
    spmm_kernel<128><<<(E + 7) / 8, 256, 0, stream>>>(
        erow, ecol, ew, bufA + (size_t)NROWS * 128, logvar, E);

    // z = mu
    hipMemcpyAsync(z, mu, (size_t)NROWS * 128 * sizeof(float),
                   hipMemcpyDeviceToDevice, stream);
}